// PoseLoss_85160611545639
// MI455X (gfx1250) — compile-verified
//
#include <hip/hip_runtime.h>
#include <math.h>

#define BATCH 512
#define NPIX  4096      // 64*64
#define NPTS  1024
#define SYMK  360

typedef __attribute__((ext_vector_type(2))) float v2f;
typedef __attribute__((ext_vector_type(8))) float v8f;

// ---- workspace layout (float offsets) ----
#define WS_GT   0                     // gt_rotation: B*9
#define WS_RS   (WS_GT + BATCH*9)     // rot_sym:     B*9
#define WS_K1   (WS_RS + BATCH*9)     // 2 blocks * 3 partials (pad to 8)
#define WS_PM   (WS_K1 + 8)           // B*4 partials
#define WS_NOCS (WS_PM + BATCH*4)     // B*2 (num, masksum)
#define WS_IVFC (WS_NOCS + BATCH*2)   // B*2

// ---------------------------------------------------------------------------
// Kernel 1: per-batch symmetry search + gt_rotation, rot_sym, small L1 terms
// ---------------------------------------------------------------------------
__global__ void k_batch(const float* __restrict__ rot,
                        const float* __restrict__ trans,
                        const float* __restrict__ size,
                        const float* __restrict__ rotation,
                        const float* __restrict__ translation,
                        const float* __restrict__ real_size,
                        const float* __restrict__ nocs_scale,
                        const int*   __restrict__ sym_info,
                        float* __restrict__ ws) {
    int b = blockIdx.x * blockDim.x + threadIdx.x;
    float r1 = 0.f, trL = 0.f, szL = 0.f;
    if (b < BATCH) {
        float G[9], P[9];
        #pragma unroll
        for (int i = 0; i < 9; i++) { G[i] = rotation[b * 9 + i]; P[i] = rot[b * 9 + i]; }

        // argmin arccos(clip(0.5*(min(tr,3)-1))) == first-occurrence argmax of clipped value
        const double q = 2.0 * M_PI / 360.0;
        float bc = 1.0f, bs = 0.0f, besth = -2.0f;
        for (int k = 0; k < SYMK; k++) {
            double th = q * (double)k;
            float c = (float)cos(th), s = (float)sin(th);
            float t = 0.f;
            #pragma unroll
            for (int i = 0; i < 3; i++) {
                float c0 = G[i*3+0]*c - G[i*3+2]*s;
                float c1 = G[i*3+1];
                float c2 = G[i*3+0]*s + G[i*3+2]*c;
                t += P[i*3+0]*c0 + P[i*3+1]*c1 + P[i*3+2]*c2;
            }
            float cv = fminf(t, 3.0f);
            float h  = 0.5f * (cv - 1.0f);
            h = fminf(fmaxf(h, -1.0f), 1.0f);
            if (h > besth) { besth = h; bc = c; bs = s; }
        }

        bool sym = (sym_info[b * 4 + 0] == 1);
        float GT[9];
        if (sym) {
            #pragma unroll
            for (int i = 0; i < 3; i++) {
                GT[i*3+0] = G[i*3+0]*bc - G[i*3+2]*bs;
                GT[i*3+1] = G[i*3+1];
                GT[i*3+2] = G[i*3+0]*bs + G[i*3+2]*bc;
            }
        } else {
            #pragma unroll
            for (int i = 0; i < 9; i++) GT[i] = G[i];
        }
        #pragma unroll
        for (int i = 0; i < 9; i++) ws[WS_GT + b*9 + i] = GT[i];

        // rot_sym[i][k] = sum_j GT[j][i] * rotation[j][k]
        #pragma unroll
        for (int i = 0; i < 3; i++)
            #pragma unroll
            for (int k = 0; k < 3; k++)
                ws[WS_RS + b*9 + i*3 + k] =
                    GT[0*3+i]*G[0*3+k] + GT[1*3+i]*G[1*3+k] + GT[2*3+i]*G[2*3+k];

        #pragma unroll
        for (int i = 0; i < 9; i++) r1 += fabsf(P[i] - GT[i]);
        float ns = nocs_scale[b];
        #pragma unroll
        for (int i = 0; i < 3; i++) {
            trL += fabsf(trans[b*3+i] - translation[b*3+i] / ns);
            szL += fabsf(size [b*3+i] - real_size  [b*3+i] / ns);
        }
    }
    __shared__ float s0[256], s1[256], s2[256];
    int t = threadIdx.x;
    s0[t] = r1; s1[t] = trL; s2[t] = szL;
    __syncthreads();
    for (int off = 128; off > 0; off >>= 1) {
        if (t < off) { s0[t] += s0[t+off]; s1[t] += s1[t+off]; s2[t] += s2[t+off]; }
        __syncthreads();
    }
    if (t == 0) {
        ws[WS_K1 + blockIdx.x*3 + 0] = s0[0];
        ws[WS_K1 + blockIdx.x*3 + 1] = s1[0];
        ws[WS_K1 + blockIdx.x*3 + 2] = s2[0];
    }
}

// ---------------------------------------------------------------------------
// Kernel 2: prop_pm term  sum |(rot-gt) @ p| over B*1024 points
// ---------------------------------------------------------------------------
__global__ void k_pm(const float* __restrict__ rot,
                     const float* __restrict__ mp,
                     const float* __restrict__ ws_gt,
                     float* __restrict__ ws_out) {
    int b = blockIdx.x >> 2;
    int p = (blockIdx.x & 3) * 256 + threadIdx.x;   // 0..1023
    float D[9];
    #pragma unroll
    for (int i = 0; i < 9; i++) D[i] = rot[b*9 + i] - ws_gt[b*9 + i];
    const float* qp = mp + ((size_t)b * NPTS + p) * 3;
    float x = qp[0], y = qp[1], z = qp[2];
    float v = fabsf(D[0]*x + D[1]*y + D[2]*z)
            + fabsf(D[3]*x + D[4]*y + D[5]*z)
            + fabsf(D[6]*x + D[7]*y + D[8]*z);
    __shared__ float sm[256];
    int t = threadIdx.x;
    sm[t] = v;
    __syncthreads();
    for (int off = 128; off > 0; off >>= 1) {
        if (t < off) sm[t] += sm[t+off];
        __syncthreads();
    }
    if (t == 0) ws_out[blockIdx.x] = sm[0];
}

// ---------------------------------------------------------------------------
// Kernel 3: fused coordinate transform + masked L1, WMMA f32 accumulator.
// D = A x ones + C  =>  sum(D)/16 == sum of all values fed through A.
// Exact (x*1.0 FMA), deterministic, layout-independent.
// ---------------------------------------------------------------------------
__global__ void k_coor(const float* __restrict__ pred,
                       const float* __restrict__ coord,
                       const float* __restrict__ mask,
                       const float* __restrict__ ws_rs,
                       float* __restrict__ ws_out) {
    int b = blockIdx.x;
    int t = threadIdx.x;
    float R[9];
    #pragma unroll
    for (int i = 0; i < 9; i++) R[i] = ws_rs[b*9 + i];

    const float* cb = coord + (size_t)b * 3 * NPIX;
    const float* pb = pred  + (size_t)b * 3 * NPIX;
    const float* mb = mask  + (size_t)b * NPIX;

    v8f accN = {};         // numerator accumulator (16x16 f32 frag)
    v8f accM = {};         // mask-sum accumulator
    v2f ones = {1.0f, 1.0f};

    #pragma unroll
    for (int it = 0; it < 8; it++) {
        int p0 = it * 256 + t;      // 0..2047
        int p1 = p0 + 2048;         // 2048..4095

        float x0 = cb[p0],        y0 = cb[NPIX + p0],  z0 = cb[2*NPIX + p0];
        float x1 = cb[p1],        y1 = cb[NPIX + p1],  z1 = cb[2*NPIX + p1];
        float m0 = mb[p0],        m1 = mb[p1];

        float g00 = R[0]*x0 + R[1]*y0 + R[2]*z0;
        float g01 = R[3]*x0 + R[4]*y0 + R[5]*z0;
        float g02 = R[6]*x0 + R[7]*y0 + R[8]*z0;
        float g10 = R[0]*x1 + R[1]*y1 + R[2]*z1;
        float g11 = R[3]*x1 + R[4]*y1 + R[5]*z1;
        float g12 = R[6]*x1 + R[7]*y1 + R[8]*z1;

        float v0 = (fabsf(pb[p0] - g00) + fabsf(pb[NPIX + p0] - g01)
                  + fabsf(pb[2*NPIX + p0] - g02)) * m0;
        float v1 = (fabsf(pb[p1] - g10) + fabsf(pb[NPIX + p1] - g11)
                  + fabsf(pb[2*NPIX + p1] - g12)) * m1;

        v2f aN = {v0, v1};
        v2f aM = {m0, m1};
        accN = __builtin_amdgcn_wmma_f32_16x16x4_f32(false, aN, false, ones,
                                                     (short)0, accN, false, false);
        accM = __builtin_amdgcn_wmma_f32_16x16x4_f32(false, aM, false, ones,
                                                     (short)0, accM, false, false);
    }

    float sN = 0.f, sM = 0.f;
    #pragma unroll
    for (int i = 0; i < 8; i++) { sN += accN[i]; sM += accM[i]; }

    __shared__ float l0[256], l1[256];
    l0[t] = sN; l1[t] = sM;
    __syncthreads();
    for (int off = 128; off > 0; off >>= 1) {
        if (t < off) { l0[t] += l0[t+off]; l1[t] += l1[t+off]; }
        __syncthreads();
    }
    if (t == 0) {
        ws_out[b*2 + 0] = l0[0] * (1.0f / 16.0f);   // each value replicated across 16 cols
        ws_out[b*2 + 1] = l1[0] * (1.0f / 16.0f);
    }
}

// ---------------------------------------------------------------------------
// Final combine: serial double-precision, fully deterministic.
// ---------------------------------------------------------------------------
__global__ void k_final(const float* __restrict__ ws, float* __restrict__ out) {
    if (blockIdx.x == 0 && threadIdx.x == 0) {
        double r1 = 0, trL = 0, szL = 0;
        for (int i = 0; i < 2; i++) {
            r1  += ws[WS_K1 + i*3 + 0];
            trL += ws[WS_K1 + i*3 + 1];
            szL += ws[WS_K1 + i*3 + 2];
        }
        double pm = 0;
        for (int i = 0; i < BATCH*4; i++) pm += ws[WS_PM + i];
        double nn = 0, nm = 0, iv = 0, im = 0;
        for (int i = 0; i < BATCH; i++) {
            nn += ws[WS_NOCS + i*2 + 0];
            nm += ws[WS_NOCS + i*2 + 1];
            iv += ws[WS_IVFC + i*2 + 0];
            im += ws[WS_IVFC + i*2 + 1];
        }
        out[0] = (float)(8.0 * r1  / (BATCH * 9.0));
        out[1] = (float)(8.0 * trL / (BATCH * 3.0));
        out[2] = (float)(8.0 * szL / (BATCH * 3.0));
        out[3] = (float)(pm / (BATCH * 3.0 * 1024.0));
        out[4] = (float)(3.0 * nn / (nm * 3.0 + 1e-8));
        out[5] = (float)(3.0 * iv / (im * 3.0 + 1e-8));
    }
}

// ---------------------------------------------------------------------------
extern "C" void kernel_launch(void* const* d_in, const int* in_sizes, int n_in,
                              void* d_out, int out_size, void* d_ws, size_t ws_size,
                              hipStream_t stream) {
    const float* rot            = (const float*)d_in[0];
    const float* trans          = (const float*)d_in[1];
    const float* size_in        = (const float*)d_in[2];
    const float* nocs_coor_pred = (const float*)d_in[3];
    const float* ivfc_coor_pred = (const float*)d_in[4];
    const float* rotation       = (const float*)d_in[5];
    const float* translation    = (const float*)d_in[6];
    const float* real_size      = (const float*)d_in[7];
    const float* roi_mask       = (const float*)d_in[8];
    const float* roi_ivfc_mask  = (const float*)d_in[9];
    const float* nocs_scale     = (const float*)d_in[10];
    const float* nocs_coord     = (const float*)d_in[11];
    const float* ivfc_coord     = (const float*)d_in[12];
    const float* model_point    = (const float*)d_in[13];
    const int*   sym_info       = (const int*)d_in[14];
    float* ws  = (float*)d_ws;
    float* out = (float*)d_out;

    k_batch<<<2, 256, 0, stream>>>(rot, trans, size_in, rotation, translation,
                                   real_size, nocs_scale, sym_info, ws);
    k_pm<<<BATCH * 4, 256, 0, stream>>>(rot, model_point, ws + WS_GT, ws + WS_PM);
    k_coor<<<BATCH, 256, 0, stream>>>(nocs_coor_pred, nocs_coord, roi_mask,
                                      ws + WS_RS, ws + WS_NOCS);
    k_coor<<<BATCH, 256, 0, stream>>>(ivfc_coor_pred, ivfc_coord, roi_ivfc_mask,
                                      ws + WS_RS, ws + WS_IVFC);
    k_final<<<1, 32, 0, stream>>>(ws, out);
}